// SpikeFP32Floor_76312978916074
// MI455X (gfx1250) — compile-verified
//
#include <hip/hip_runtime.h>
#include <stdint.h>

// ---------------------------------------------------------------------------
// SpikeFP32Floor on MI455X (gfx1250)
//  - 1 GB streaming traffic -> pure bandwidth problem (~43us @ 23.3 TB/s)
//  - wave32 ballot = single-instruction 32-bit pack (lane j holds bit j)
//  - floor() bit math runs wave-uniform on the SALU
//  - input staged via gfx1250 async global->LDS copies (ASYNCcnt), double
//    buffered per wave; LDS reads via explicit AS(3) pointers -> ds_load_b32
//  - non-temporal stores (1GB >> 192MB L2)
// ---------------------------------------------------------------------------

#define WAVES_PER_BLOCK 8
#define BLOCK_THREADS   256
#define VPW             8   // values per wave per pipeline stage (VPW*128B = 1KB)

#define HAS_ASYNC_LDS (__has_builtin(__builtin_amdgcn_global_load_async_to_lds_b128) && \
                       __has_builtin(__builtin_amdgcn_s_wait_asynccnt))

// floor() of an FP32 number given its raw bit pattern (matches reference math)
__device__ __forceinline__ uint32_t floor_fp32_bits(uint32_t u) {
  const uint32_t e = (u >> 23) & 0xFFu;   // biased exponent
  const uint32_t s = u >> 31;             // sign
  int shi = 150 - (int)e;                 // fractional mantissa bits
  shi = shi < 0 ? 0 : shi;
  shi = shi > 31 ? 31 : shi;
  const uint32_t shift = (uint32_t)shi;
  const uint32_t mask  = (1u << shift) - 1u;          // shift in [0,31]
  const uint32_t mant  = u & 0x7FFFFFu;
  const bool has_frac  = ((u & mask) & 0x7FFFFFu) != 0u;
  const uint32_t trunc = u & ~mask;                   // truncate toward zero
  // negative-with-fraction branch: -(V+1) as exact fp32 bits
  const uint32_t V = (mant | 0x800000u) >> shift;     // integer magnitude
  const uint32_t W = V + 1u;                          // in [1, 2^23] when live
  const uint32_t p = 31u - (uint32_t)__clz((int)W);   // floor(log2(W))
  const uint32_t q = (p > 23u) ? 0u : (23u - p);      // keep shift defined
  const uint32_t tm1 = 0x80000000u | ((127u + p) << 23) | ((W << q) & 0x7FFFFFu);

  uint32_t res = (s != 0u && has_frac) ? tm1 : trunc; // neg_frac_mux
  if (e < 127u)  res = s ? 0xBF800000u : 0u;          // |x| < 1 -> -1.0 or +0.0
  if (e >= 150u) res = u;                             // already integer (or NaN/Inf)
  return res;
}

// one value: lane j holds input bit j (MSB-first). ballot packs, brev reorders.
__device__ __forceinline__ void process_one(unsigned lane, float fbit,
                                            float* __restrict__ outp) {
#if __has_builtin(__builtin_amdgcn_ballot_w32)
  uint32_t b = __builtin_amdgcn_ballot_w32(fbit != 0.0f);
#else
  uint32_t b = (uint32_t)__ballot(fbit != 0.0f);
#endif
  uint32_t u   = __brev(b);               // lane j -> bit (31-j): MSB-first pack
  uint32_t res = floor_fp32_bits(u);      // wave-uniform -> SALU
  uint32_t r   = __brev(res);             // bit j of output = bit j of brev(res)
  float ov = (float)((r >> lane) & 1u);
  __builtin_nontemporal_store(ov, outp + lane);
}

#if HAS_ASYNC_LDS
typedef int v4i_b128 __attribute__((vector_size(16)));
typedef __attribute__((address_space(3))) float lds_float;
typedef __attribute__((address_space(3))) v4i_b128 lds_v4i;
typedef __attribute__((address_space(1))) v4i_b128 glb_v4i;

// stage VPW consecutive values (VPW*128B) into an LDS buffer:
// 2 async b128 copies of 512B each (32 lanes x 16B), tracked on ASYNCcnt.
__device__ __forceinline__ void async_issue(const float* __restrict__ x, unsigned v0,
                                            lds_float* lbase, unsigned lane) {
  const float* g = x + (size_t)v0 * 32u + lane * 4u;
  lds_float*   l = lbase + lane * 4u;
#pragma unroll
  for (int c = 0; c < 2; ++c) {
    __builtin_amdgcn_global_load_async_to_lds_b128(
        (glb_v4i*)(g + c * 128),
        (lds_v4i*)(l + c * 128),
        0, 0);
  }
}
#endif

__global__ void __launch_bounds__(BLOCK_THREADS)
spike_floor_kernel(const float* __restrict__ x, float* __restrict__ y, unsigned nvals) {
  const unsigned lane   = threadIdx.x & 31u;
  const unsigned wiv    = threadIdx.x >> 5;
  const unsigned gwave  = blockIdx.x * WAVES_PER_BLOCK + wiv;
  const unsigned stride = gridDim.x * WAVES_PER_BLOCK * VPW;

  unsigned base = gwave * VPW;
  if (base >= nvals) return;

#if HAS_ASYNC_LDS
  __shared__ float smem[2][WAVES_PER_BLOCK][VPW * 32];
  // explicit AS(3) pointers to this wave's two staging buffers -> ds_load path
  lds_float* lb0 = (lds_float*)&smem[0][wiv][0];
  lds_float* lb1 = (lds_float*)&smem[1][wiv][0];

  unsigned cur = 0;
  if (base + VPW <= nvals) async_issue(x, base, lb0, lane);

  for (; base < nvals; base += stride) {
    const unsigned nb   = base + stride;
    const bool curfull  = (base + VPW) <= nvals;
    const bool nxtfull  = (nb < nvals) && ((nb + VPW) <= nvals);

    if (nxtfull) async_issue(x, nb, (cur ? lb0 : lb1), lane);

    if (curfull) {
      // FIFO async completion: waiting to depth 2 (next chunk's copies)
      // guarantees the current chunk is resident.
      if (nxtfull) __builtin_amdgcn_s_wait_asynccnt(2);
      else         __builtin_amdgcn_s_wait_asynccnt(0);
#pragma unroll
      for (int k = 0; k < VPW; ++k) {
        // AS(3) load -> ds_load_b32, conflict-free (consecutive lanes/banks)
        float f = cur ? lb1[k * 32 + (int)lane] : lb0[k * 32 + (int)lane];
        process_one(lane, f, y + (size_t)(base + k) * 32u);
      }
    } else {
      // partial tail chunk: direct non-temporal loads
      for (unsigned v = base; v < nvals; ++v) {
        float f = __builtin_nontemporal_load(&x[(size_t)v * 32u + lane]);
        process_one(lane, f, y + (size_t)v * 32u);
      }
    }
    cur ^= 1u;
  }
#else
  // fallback: direct non-temporal streaming loads (still fully coalesced)
  for (; base < nvals; base += stride) {
#pragma unroll
    for (int k = 0; k < VPW; ++k) {
      unsigned v = base + k;
      if (v < nvals) {
        float f = __builtin_nontemporal_load(&x[(size_t)v * 32u + lane]);
        process_one(lane, f, y + (size_t)v * 32u);
      }
    }
  }
#endif
}

extern "C" void kernel_launch(void* const* d_in, const int* in_sizes, int n_in,
                              void* d_out, int out_size, void* d_ws, size_t ws_size,
                              hipStream_t stream) {
  (void)n_in; (void)out_size; (void)d_ws; (void)ws_size;
  const float* x = (const float*)d_in[0];
  float*       y = (float*)d_out;
  const unsigned nvals = (unsigned)(in_sizes[0] / 32);  // 4,194,304

  unsigned waves_needed = (nvals + VPW - 1) / VPW;
  unsigned blocks = (waves_needed + WAVES_PER_BLOCK - 1) / WAVES_PER_BLOCK;
  if (blocks > 2048u) blocks = 2048u;   // grid-stride covers the rest
  if (blocks == 0u)   blocks = 1u;

  spike_floor_kernel<<<dim3(blocks), dim3(BLOCK_THREADS), 0, stream>>>(x, y, nvals);
}